// SpatialMaxpool2d_8409545965923
// MI455X (gfx1250) — compile-verified
//
#include <hip/hip_runtime.h>
#include <cstdint>

// Problem shape (fixed by the reference).
#define BB 64
#define CC 133
#define CM 128
#define HH 64
#define WW 64
#define HO 32
#define WO 32

#define NBUF 16        // async pipeline depth (channels in flight per wave)
#define TPB  256       // 8 wave32s per block; one wave per output row

// Copy 16 bytes global -> LDS via the CDNA5 async DMA path (ASYNCcnt-tracked).
// Inline asm: VDST = LDS byte address VGPR, VADDR = 64-bit global address pair.
__device__ __forceinline__ void async_ld16(const float* g, float* l) {
  asm volatile("global_load_async_to_lds_b128 %0, %1, off"
               :: "v"((uint32_t)(uintptr_t)l),
                  "v"((uint64_t)(uintptr_t)g)
               : "memory");
}

#define WAIT_ASYNC(n) asm volatile("s_wait_asynccnt %0" :: "i"(n) : "memory")
#define WAIT_DS0()    asm volatile("s_wait_dscnt 0" ::: "memory")

extern "C" __global__ void __launch_bounds__(TPB)
pool_stats_kernel(const float* __restrict__ x, float* __restrict__ out) {
  // 16 circular slabs of 4KB: slab = 16 contiguous input rows for one channel.
  __shared__ __align__(16) float lds[NBUF * 1024];

  const int tid  = threadIdx.x;
  const int wv   = tid >> 5;          // wave id 0..7
  const int ln   = tid & 31;          // lane  = wo (Wo == 32 == wave32)
  const int b    = blockIdx.x >> 2;   // batch
  const int tile = blockIdx.x & 3;    // 4 row-tiles of 8 output rows
  const int h0   = tile << 4;         // first input row of slab (16 rows)
  const int ho   = (tile << 3) + wv;  // this wave's output row

  // Per-thread global source offset inside a channel slab (16B each).
  const size_t gslab = ((size_t)b * CC * HH + h0) * (size_t)WW + (size_t)(tid << 2);
  float* lbase = lds + (tid << 2);

  // LDS float2 read indices for this wave's two input rows (conflict-free).
  const int i0 = (wv << 6) + ln;      // row 2*wv   : float2 index
  const int i1 = i0 + 32;             // row 2*wv+1

  float* outp = out + ((size_t)b * CC * HO + ho) * (size_t)WO + ln;

  auto issue = [&](int c) {
    const int slot = c & (NBUF - 1);
    async_ld16(x + gslab + (size_t)c * (HH * WW), lbase + slot * 1024);
  };

  unsigned cnt = 0;  // 4 packed byte counters: argmax histogram (max 128/byte)

  auto process = [&](int c) {
    const int slot = c & (NBUF - 1);
    const float2* l2 = reinterpret_cast<const float2*>(lds) + slot * 512;
    const float2 r0 = l2[i0];
    const float2 r1 = l2[i1];
    float m = r0.x; unsigned idx = 0u;            // first-max tie-break,
    if (r0.y > m) { m = r0.y; idx = 1u; }         // matches jnp.argmax
    if (r1.x > m) { m = r1.x; idx = 2u; }
    if (r1.y > m) { m = r1.y; idx = 3u; }
    cnt += 1u << (idx << 3);
    outp[(size_t)c * (HO * WO)] = m;              // pooled channel c
  };

  // ---- software pipeline over the 128 pooled channels ----
#pragma unroll
  for (int k = 0; k < NBUF; ++k) issue(k);

  for (int c = 0; c < CM - NBUF; ++c) {
    WAIT_ASYNC(NBUF - 1);   // oldest outstanding slab (channel c) has landed
    process(c);
    WAIT_DS0();             // our ds reads done -> safe to overwrite the slot
    issue(c + NBUF);
  }
  WAIT_ASYNC(0);            // drain tail
  for (int c = CM - NBUF; c < CM; ++c) process(c);

  // ---- statistics from channels 128..132 (direct global loads) ----
  const float inv = 1.0f / 128.0f;
  const float w0 = (float)(cnt        & 255u) * inv;
  const float w1 = (float)((cnt >> 8) & 255u) * inv;
  const float w2 = (float)((cnt >> 16)& 255u) * inv;
  const float w3 = (float)(cnt >> 24)         * inv;

  auto dot4 = [&](float a0, float a1, float a2, float a3) {
    return fmaf(a0, w0, fmaf(a1, w1, fmaf(a2, w2, a3 * w3)));
  };

  const float* sp = x + (((size_t)b * CC + CM) * HH + 2 * ho) * (size_t)WW + 2 * ln;
  const int CS = HH * WW;  // channel stride (4096 floats)

  const float2 m0a = *(const float2*)(sp);                 // ch128 row0
  const float2 m0b = *(const float2*)(sp + WW);            // ch128 row1
  const float2 m1a = *(const float2*)(sp + CS);            // ch129
  const float2 m1b = *(const float2*)(sp + CS + WW);
  const float2 v0a = *(const float2*)(sp + 2 * CS);        // ch130
  const float2 v0b = *(const float2*)(sp + 2 * CS + WW);
  const float2 v1a = *(const float2*)(sp + 3 * CS);        // ch131
  const float2 v1b = *(const float2*)(sp + 3 * CS + WW);
  const float2 cva = *(const float2*)(sp + 4 * CS);        // ch132
  const float2 cvb = *(const float2*)(sp + 4 * CS + WW);

  const float mean0 = dot4(m0a.x, m0a.y, m0b.x, m0b.y);
  const float mean1 = dot4(m1a.x, m1a.y, m1b.x, m1b.y);
  const float var1_0 = dot4(v0a.x, v0a.y, v0b.x, v0b.y);
  const float var1_1 = dot4(v1a.x, v1a.y, v1b.x, v1b.y);
  const float cov1   = dot4(cva.x, cva.y, cvb.x, cvb.y);

  const float d00 = m0a.x - mean0, d01 = m0a.y - mean0;
  const float d02 = m0b.x - mean0, d03 = m0b.y - mean0;
  const float d10 = m1a.x - mean1, d11 = m1a.y - mean1;
  const float d12 = m1b.x - mean1, d13 = m1b.y - mean1;

  const float var2_0 = dot4(d00 * d00, d01 * d01, d02 * d02, d03 * d03);
  const float var2_1 = dot4(d10 * d10, d11 * d11, d12 * d12, d13 * d13);
  const float cov2   = dot4(d00 * d10, d01 * d11, d02 * d12, d03 * d13);

  float* so = outp + (size_t)CM * (HO * WO);
  so[0 * (HO * WO)] = mean0;
  so[1 * (HO * WO)] = mean1;
  so[2 * (HO * WO)] = var1_0 + var2_0;
  so[3 * (HO * WO)] = var1_1 + var2_1;
  so[4 * (HO * WO)] = cov1 + cov2;
}

extern "C" void kernel_launch(void* const* d_in, const int* in_sizes, int n_in,
                              void* d_out, int out_size, void* d_ws, size_t ws_size,
                              hipStream_t stream) {
  (void)in_sizes; (void)n_in; (void)out_size; (void)d_ws; (void)ws_size;
  const float* x = (const float*)d_in[0];
  float* out = (float*)d_out;
  dim3 grid(BB * (HO / 8));   // 256 blocks: (batch, 8-row tile)
  dim3 block(TPB);            // 8 wave32s, one per output row
  hipLaunchKernelGGL(pool_stats_kernel, grid, block, 0, stream, x, out);
}